// PhysicsConstrainedLoss_27891517620453
// MI455X (gfx1250) — compile-verified
//
#include <hip/hip_runtime.h>
#include <hip/hip_bf16.h>
#include <math.h>

#define B_ROWS   1048576
#define N_WL     512
#define NBLK     1024
#define BLOCK    256
#define NACC     12

typedef __attribute__((ext_vector_type(2))) float v2f;
typedef __attribute__((ext_vector_type(8))) float v8f;

// ---------------------------------------------------------------------------
// Main streaming kernel: CE + argmax (wave-cooperative) and scalar-stream
// sufficient statistics. Writes 12 partial sums per block to ws, transposed
// as ws[q*NBLK + blockIdx.x] so the finalize WMMA stage reads contiguously.
// ---------------------------------------------------------------------------
__global__ __launch_bounds__(BLOCK)
void pcl_main(const float* __restrict__ logits,
              const int*   __restrict__ labels,
              const float* __restrict__ reg,
              const float* __restrict__ normc,
              const float* __restrict__ origc,
              const float* __restrict__ attn,
              float* __restrict__ ws)
{
    const int tid      = blockIdx.x * blockDim.x + threadIdx.x;
    const int lane     = threadIdx.x & 31;
    const int wid      = threadIdx.x >> 5;
    const int nthreads = gridDim.x * blockDim.x;
    const int nwaves   = nthreads >> 5;
    const int gwave    = tid >> 5;

    float acc[NACC];
#pragma unroll
    for (int q = 0; q < NACC; ++q) acc[q] = 0.f;

    // ---- Loop A: 4 rows per wave-iteration; each lane owns 4 columns ----
    const float4* lg4    = (const float4*)logits;
    const int ngroups    = B_ROWS / 4;
    const int subrow     = lane >> 3;        // which of the 4 rows
    const int c0         = (lane & 7) * 4;   // first column this lane covers
    const int gleader    = lane & ~7;        // leader lane of 8-lane row group

    for (int g = gwave; g < ngroups; g += nwaves) {
        float4 v = lg4[(size_t)g * 32 + lane];
        if (g + nwaves < ngroups)
            __builtin_prefetch(&lg4[(size_t)(g + nwaves) * 32 + lane], 0, 1);
        int row = g * 4 + subrow;

        // first-occurrence max/argmax over this lane's 4 values
        float mx = v.x; int mi = c0;
        if (v.y > mx) { mx = v.y; mi = c0 + 1; }
        if (v.z > mx) { mx = v.z; mi = c0 + 2; }
        if (v.w > mx) { mx = v.w; mi = c0 + 3; }
        // reduce (max, first-index) across the 8-lane row group
#pragma unroll
        for (int m = 1; m <= 4; m <<= 1) {
            float omx = __shfl_xor(mx, m, 32);
            int   omi = __shfl_xor(mi, m, 32);
            if (omx > mx || (omx == mx && omi < mi)) { mx = omx; mi = omi; }
        }
        // log-sum-exp
        float s = __expf(v.x - mx) + __expf(v.y - mx) +
                  __expf(v.z - mx) + __expf(v.w - mx);
#pragma unroll
        for (int m = 1; m <= 4; m <<= 1) s += __shfl_xor(s, m, 32);
        float lse = __logf(s);

        int lab = labels[row];   // uniform within the 8-lane group
        if (lab >= c0 && lab < c0 + 4) {
            float a4[4] = {v.x, v.y, v.z, v.w};
            acc[0] += -(a4[lab - c0] - mx - lse);          // cross-entropy
        }
        if (lane == gleader) {
            float rr = reg[row];
            float p  = (float)mi;                          // argmax as float
            acc[1] += p;
            acc[2] += p * p;
            acc[3] += p * rr;
        }
    }

    // ---- Loop B: per-thread coalesced scalar streams ----
    for (int i = tid; i < B_ROWS; i += nthreads) {
        float rg = reg[i];
        float d  = rg - normc[i];
        acc[4]  += d * d;          // regression MSE numerator
        acc[5]  += rg;             // sum r
        acc[6]  += rg * rg;        // sum r^2
        float x  = log1pf(origc[i]);
        float y  = attn[i];
        acc[7]  += x;
        acc[8]  += y;
        acc[9]  += x * x;
        acc[10] += y * y;
        acc[11] += x * y;
    }

    // ---- block reduction: wave shuffle -> LDS -> 12 stores ----
    __shared__ float sred[BLOCK / 32][NACC];
#pragma unroll
    for (int q = 0; q < NACC; ++q) {
        float v = acc[q];
#pragma unroll
        for (int m = 1; m < 32; m <<= 1) v += __shfl_xor(v, m, 32);
        if (lane == 0) sred[wid][q] = v;
    }
    __syncthreads();
    if (threadIdx.x < NACC) {
        float v = 0.f;
#pragma unroll
        for (int w = 0; w < BLOCK / 32; ++w) v += sred[w][threadIdx.x];
        ws[threadIdx.x * NBLK + blockIdx.x] = v;   // transposed for WMMA stage
    }
}

// ---------------------------------------------------------------------------
// Finalize: one wave. Spectral/peak terms over 512 wavelengths, then the
// 12 x 1024 partial sums reduced with chained V_WMMA_F32_16X16X4_F32
// (A = ones(16x4), B = 64 partials per call, C accumulates).
// ---------------------------------------------------------------------------
__global__ __launch_bounds__(32)
void pcl_finalize(const float* __restrict__ ws,
                  const float* __restrict__ aw,
                  const float* __restrict__ rc,
                  float* __restrict__ out)
{
    const int lane = threadIdx.x;   // blockDim.x == 32, EXEC all ones

    // ---- spectral + peak-mask terms over N_WL = 512 ----
    float dot = 0.f, w2 = 0.f, r2 = 0.f, pk = 0.f;
    const int peaks[9] = {1680, 1695, 1705, 1725, 1730, 1760, 1765, 1780, 1790};
    for (int i = lane; i < N_WL; i += 32) {
        float w = aw[i], r = rc[i];
        dot += w * r; w2 += w * w; r2 += r * r;
        float m = 0.f;
#pragma unroll
        for (int p = 0; p < 9; ++p) {
            int idx = (int)((double)(peaks[p] - 889) / (1710.0 - 889.0) * 512.0);
            if (i >= idx - 10 && i <= idx + 10) {
                float t = (float)(i - idx) * 0.2f;
                m = fmaxf(m, __expf(-0.5f * t * t));
            }
        }
        float d = w * m - r * m;
        pk += d * d;
    }
#pragma unroll
    for (int m = 1; m < 32; m <<= 1) {
        dot += __shfl_xor(dot, m, 32);
        w2  += __shfl_xor(w2,  m, 32);
        r2  += __shfl_xor(r2,  m, 32);
        pk  += __shfl_xor(pk,  m, 32);
    }

    // ---- WMMA reduction of the 12 streams of NBLK per-block partials ----
    v2f a; a[0] = 1.f; a[1] = 1.f;          // A = ones(16x4)
    float S[NACC];
#pragma unroll
    for (int q = 0; q < NACC; ++q) {
        const float* col = ws + q * NBLK;
        v8f c = {};                          // C/D accumulator (16x16 f32)
        for (int base = 0; base < NBLK; base += 64) {
            v2f b;                           // 64 partials -> B(4x16)
            b[0] = col[base + 2 * lane];
            b[1] = col[base + 2 * lane + 1];
            // D[m][n] = sum_k B[k][n] + C[m][n]; row 0 accumulates col-sums
            c = __builtin_amdgcn_wmma_f32_16x16x4_f32(
                    false, a, false, b, (short)0, c, false, false);
        }
        // D row 0 lives in c[0], lanes 0..15: fold to the total
        float t = c[0];
#pragma unroll
        for (int m = 1; m < 16; m <<= 1) t += __shfl_xor(t, m, 32);
        S[q] = __shfl(t, 0, 32);             // broadcast, no EXEC divergence
    }

    if (lane == 0) {
        const float Bf  = (float)B_ROWS;
        const float eps = 1e-8f;
        float loss_cls = S[0] / Bf;
        float loss_reg = S[4] / Bf;

        float nw = fmaxf(sqrtf(w2), eps);
        float nr = fmaxf(sqrtf(r2), eps);
        float spectral = 1.f - dot / (nw * nr);
        float peaksL   = pk / (float)N_WL;

        // concentration correlation: x = log1p(orig), y = attention
        float sx = S[7], sy = S[8], sxx = S[9], syy = S[10], sxy = S[11];
        float num  = sxy - sx * sy / Bf;
        float den  = sqrtf((sxx - sx * sx / Bf) * (syy - sy * sy / Bf));
        float rco  = num / den;
        float conc = (rco != rco) ? 0.f : 1.f - fabsf(rco);

        // consistency correlation: p = argmax class, r = regression
        float sp = S[1], sr = S[5], spp = S[2], srr = S[6], spr = S[3];
        float num2 = spr - sp * sr / Bf;
        float den2 = sqrtf((spp - sp * sp / Bf) * (srr - sr * sr / Bf));
        float rcs  = num2 / den2;
        float cons = (rcs != rcs) ? 0.f : 1.f - fabsf(rcs);

        float phys = 0.3f * spectral + 0.3f * peaksL + 0.2f * conc + 0.2f * cons;
        out[0] = 0.45f * loss_cls + 0.45f * loss_reg + 0.1f * phys;
    }
}

// ---------------------------------------------------------------------------
extern "C" void kernel_launch(void* const* d_in, const int* in_sizes, int n_in,
                              void* d_out, int out_size, void* d_ws, size_t ws_size,
                              hipStream_t stream)
{
    const float* logits = (const float*)d_in[0];
    const int*   labels = (const int*)  d_in[1];
    const float* reg    = (const float*)d_in[2];
    const float* normc  = (const float*)d_in[3];
    const float* origc  = (const float*)d_in[4];
    const float* attn   = (const float*)d_in[5];
    const float* aw     = (const float*)d_in[6];
    const float* rc     = (const float*)d_in[7];
    float* ws  = (float*)d_ws;   // needs 12 * NBLK * 4 = 48 KB
    float* out = (float*)d_out;

    pcl_main<<<NBLK, BLOCK, 0, stream>>>(logits, labels, reg, normc, origc, attn, ws);
    pcl_finalize<<<1, 32, 0, stream>>>(ws, aw, rc, out);
}